// TopKSoftMax_56392920597026
// MI455X (gfx1250) — compile-verified
//
#include <hip/hip_runtime.h>
#include <cstdint>

// Descending compare-exchange: a <- max, b <- min
#define CE(a, b)                          \
  do {                                    \
    float _hi = fmaxf((a), (b));          \
    float _lo = fminf((a), (b));          \
    (a) = _hi;                            \
    (b) = _lo;                            \
  } while (0)

// CDNA5 async copy: 16 bytes per lane, global -> LDS, tracked by ASYNCcnt.
__device__ __forceinline__ void async_ld16(uint32_t lds_off, const float* gaddr) {
  asm volatile("global_load_async_to_lds_b128 %0, %1, off"
               :
               : "v"(lds_off), "v"(gaddr)
               : "memory");
}

// One wave32 handles 4 rows per iteration: 8 lanes per row, 8 elements per lane.
__global__ __launch_bounds__(256) void TopKSoftMax_kernel(
    const float* __restrict__ in, float* __restrict__ out, int nGroups) {
  // [buf][wave][256 floats] : 2 * 8 * 1KB = 16KB per block
  __shared__ float lds[2 * 8 * 256];

  const int tid = threadIdx.x;
  const int wave = tid >> 5;
  const int lane = tid & 31;
  const int wavesPerBlock = blockDim.x >> 5;
  const int stride = gridDim.x * wavesPerBlock;
  int g = blockIdx.x * wavesPerBlock + wave;

  // LDS staging bases (low 32 bits of a flat shared pointer == LDS offset)
  const uint32_t ldsByte0 = (uint32_t)(uintptr_t)(&lds[(0 * 8 + wave) * 256]);
  const uint32_t ldsByte1 = (uint32_t)(uintptr_t)(&lds[(1 * 8 + wave) * 256]);

  bool valid = g < nGroups;
  int buf = 0;

  // Prologue: issue first group's async loads into buffer 0.
  if (valid) {
    const float* gp = in + (size_t)g * 256 + lane * 4;
    async_ld16(ldsByte0 + lane * 16, gp);
    async_ld16(ldsByte0 + 512 + lane * 16, gp + 128);
  }

  while (valid) {
    const int gn = g + stride;
    const bool vn = gn < nGroups;

    // Issue next group's loads into the other buffer (double buffering).
    if (vn) {
      const uint32_t nb = (buf == 0) ? ldsByte1 : ldsByte0;
      const float* gp = in + (size_t)gn * 256 + lane * 4;
      async_ld16(nb + lane * 16, gp);
      async_ld16(nb + 512 + lane * 16, gp + 128);
      // 2 outstanding = the next buffer; current buffer is complete (in-order).
      asm volatile("s_wait_asynccnt 0x2" ::: "memory");
    } else {
      asm volatile("s_wait_asynccnt 0x0" ::: "memory");
    }

    // ---- load this lane's 8 elements from LDS (32B contiguous per lane) ----
    const int ldsFloatBase = (buf * 8 + wave) * 256 + lane * 8;
    const float4 va = *reinterpret_cast<const float4*>(&lds[ldsFloatBase]);
    const float4 vb = *reinterpret_cast<const float4*>(&lds[ldsFloatBase + 4]);

    float x[8];
    x[0] = va.x; x[1] = va.y; x[2] = va.z; x[3] = va.w;
    x[4] = vb.x; x[5] = vb.y; x[6] = vb.z; x[7] = vb.w;

    // ---- local sort (descending), Batcher odd-even merge sort, 19 CEs ----
    float s[8];
#pragma unroll
    for (int i = 0; i < 8; ++i) s[i] = x[i];
    CE(s[0], s[1]); CE(s[2], s[3]); CE(s[4], s[5]); CE(s[6], s[7]);
    CE(s[0], s[2]); CE(s[1], s[3]); CE(s[4], s[6]); CE(s[5], s[7]);
    CE(s[1], s[2]); CE(s[5], s[6]);
    CE(s[0], s[4]); CE(s[1], s[5]); CE(s[2], s[6]); CE(s[3], s[7]);
    CE(s[2], s[4]); CE(s[3], s[5]);
    CE(s[1], s[2]); CE(s[3], s[4]); CE(s[5], s[6]);

    // ---- 3 cross-lane merge levels within the 8-lane row group ----
#pragma unroll
    for (int m = 1; m <= 4; m <<= 1) {
      float r[8];
#pragma unroll
      for (int i = 0; i < 8; ++i) r[i] = __shfl_xor(s[i], m, 32);
      float t[8];
#pragma unroll
      for (int i = 0; i < 8; ++i) t[i] = fmaxf(s[i], r[7 - i]);
      // bitonic clean-up, descending
      CE(t[0], t[4]); CE(t[1], t[5]); CE(t[2], t[6]); CE(t[3], t[7]);
      CE(t[0], t[2]); CE(t[1], t[3]); CE(t[4], t[6]); CE(t[5], t[7]);
      CE(t[0], t[1]); CE(t[2], t[3]); CE(t[4], t[5]); CE(t[6], t[7]);
#pragma unroll
      for (int i = 0; i < 8; ++i) s[i] = t[i];
    }

    const float rowMax = s[0];
    const float thresh = s[7];   // 8th largest of the 64-element row

    // ---- masked softmax ----
    float p[8];
    float lsum = 0.0f;
#pragma unroll
    for (int i = 0; i < 8; ++i) {
      float e = __expf(x[i] - rowMax);
      p[i] = (x[i] >= thresh) ? e : 0.0f;
      lsum += p[i];
    }
    lsum += __shfl_xor(lsum, 1, 32);
    lsum += __shfl_xor(lsum, 2, 32);
    lsum += __shfl_xor(lsum, 4, 32);
    const float inv = 1.0f / lsum;

    float4 oa, ob;
    oa.x = p[0] * inv; oa.y = p[1] * inv; oa.z = p[2] * inv; oa.w = p[3] * inv;
    ob.x = p[4] * inv; ob.y = p[5] * inv; ob.z = p[6] * inv; ob.w = p[7] * inv;

    float* op = out + (size_t)g * 256 + lane * 8;
    *reinterpret_cast<float4*>(op) = oa;
    *reinterpret_cast<float4*>(op + 4) = ob;

    g = gn;
    buf ^= 1;
    valid = vn;
  }
}

extern "C" void kernel_launch(void* const* d_in, const int* in_sizes, int n_in,
                              void* d_out, int out_size, void* d_ws, size_t ws_size,
                              hipStream_t stream) {
  (void)n_in; (void)out_size; (void)d_ws; (void)ws_size;
  const float* in = (const float*)d_in[0];
  float* out = (float*)d_out;

  const long long totalElems = (long long)in_sizes[0];
  const int rows = (int)(totalElems / 64);
  const int nGroups = rows >> 2;  // 4 rows per wave-iteration

  const int threads = 256;                    // 8 waves
  const int wavesPerBlock = threads / 32;
  int blocks = 4096;                          // grid-stride, ~8 groups per wave
  int needed = (nGroups + wavesPerBlock - 1) / wavesPerBlock;
  if (blocks > needed) blocks = needed;
  if (blocks < 1) blocks = 1;

  TopKSoftMax_kernel<<<blocks, threads, 0, stream>>>(in, out, nGroups);
}